// MultiHeadedIGAT_28819230556369
// MI455X (gfx1250) — compile-verified
//
#include <hip/hip_runtime.h>
#include <hip/hip_bf16.h>
#include <math.h>

// Problem constants (fixed by the reference)
#define F_DIM 128   // input feature dim
#define H_DIM 4     // heads
#define A_DIM 32    // attention size
#define O_DIM 64    // out size per head
#define HO    256   // H_DIM * O_DIM
#define CG_ITERS 48 // fixed CG iteration count (kappa ~ 16 => converges << 48)

typedef float v2f __attribute__((ext_vector_type(2)));
typedef float v8f __attribute__((ext_vector_type(8)));

#define AS_LDS __attribute__((address_space(3)))
#define AS_GLB __attribute__((address_space(1)))

#if __has_builtin(__builtin_amdgcn_global_load_async_to_lds_b32) && \
    __has_builtin(__builtin_amdgcn_s_wait_asynccnt)
#define HAVE_ASYNC_LDS 1
#else
#define HAVE_ASYNC_LDS 0
#endif

__device__ __forceinline__ float sigmoidf_(float x) {
  return 1.0f / (1.0f + __expf(-x));
}

// ---------------------------------------------------------------------------
// GEMM: Y[N,ncols] = X[N,128] @ W[128,ncols] + bias, via V_WMMA_F32_16X16X4_F32
// Block = 256 threads (8 waves). Block owns a 128-row x 16-col output panel;
// each wave computes one 16x16 tile. The shared 128x16 B tile is staged once
// into LDS (async-to-LDS path on CDNA5, ASYNCcnt tracked), then all 8 waves
// feed WMMA from ds_load -- B global traffic drops 8x and stays coalesced.
// A layout (16x4 f32, 2 VGPRs): lane holds row M=lane&15; lanes 0-15 hold
//   K={k0,k0+1}, lanes 16-31 hold K={k0+2,k0+3}. B: col N=lane&15, same K
//   split. D: vgpr r, lane l -> row r+8*(l>>4), col l&15.
// ---------------------------------------------------------------------------
__global__ __launch_bounds__(256) void igat_gemm_wmma(
    const float* __restrict__ X, const float* __restrict__ W,
    const float* __restrict__ bias, float* __restrict__ Y,
    int nrows, int ncols) {
  __shared__ float ldsB[F_DIM * 16];  // 8 KB B tile: [k][n]

  const int tid  = threadIdx.x;
  const int lane = tid & 31;
  const int wave = tid >> 5;
  const int col0 = blockIdx.y * 16;

  // ---- Stage B tile (W[0:128, col0:col0+16]) into LDS, coalesced ----
  {
    const int kk = tid >> 4;  // 0..15
    const int nn = tid & 15;
#if HAVE_ASYNC_LDS
#pragma unroll
    for (int k = kk; k < F_DIM; k += 16) {
      __builtin_amdgcn_global_load_async_to_lds_b32(
          (AS_GLB int*)(W + (size_t)k * ncols + col0 + nn),
          (AS_LDS int*)(&ldsB[k * 16 + nn]), /*offset=*/0, /*cpol=*/0);
    }
    __builtin_amdgcn_s_wait_asynccnt(0);
    __syncthreads();
#else
#pragma unroll
    for (int k = kk; k < F_DIM; k += 16)
      ldsB[k * 16 + nn] = W[(size_t)k * ncols + col0 + nn];
    __syncthreads();
#endif
  }

  const int m     = lane & 15;  // A-row within tile
  const int khalf = lane >> 4;  // 0 -> K pair {0,1}; 1 -> K pair {2,3}
  const int n     = lane & 15;  // B/D column

  // Clamp row for loads so EXEC stays all-1s for WMMA (stores are guarded).
  int arow = blockIdx.x * 128 + wave * 16 + m;
  if (arow >= nrows) arow = nrows - 1;
  const float* __restrict__ xrow = X + (size_t)arow * F_DIM;

  v8f acc = {};
#pragma unroll
  for (int k0 = 0; k0 < F_DIM; k0 += 4) {
    const int ka = k0 + 2 * khalf;
    v2f av = *(const v2f*)(xrow + ka);  // 8B-aligned global_load_b64
    v2f bv;
    bv.x = ldsB[(ka + 0) * 16 + n];     // conflict-free ds_load
    bv.y = ldsB[(ka + 1) * 16 + n];
    acc = __builtin_amdgcn_wmma_f32_16x16x4_f32(
        /*neg_a=*/false, av, /*neg_b=*/false, bv,
        /*c_mod=*/(short)0, acc, /*reuse_a=*/false, /*reuse_b=*/false);
  }

  const float bval = bias[col0 + n];
  const int rbase = blockIdx.x * 128 + wave * 16;
#pragma unroll
  for (int rr = 0; rr < 8; ++rr) {
    const int mrow = rbase + rr + 8 * khalf;
    if (mrow < nrows)
      Y[(size_t)mrow * ncols + col0 + n] = acc[rr] + bval;
  }
}

// ---------------------------------------------------------------------------
// Per-edge attention: attn[e,h] = sigmoid(<a[row[e],h,:], a[col[e],h,:]>),
// plus deg[row[e],h] += attn (edge list is already symmetrized).
// One wave per edge; head h occupies lanes over a[.., h*32 + lane].
// ---------------------------------------------------------------------------
__global__ void igat_edge_attn(const int* __restrict__ row,
                               const int* __restrict__ col,
                               const float* __restrict__ a,
                               float* __restrict__ attn,
                               float* __restrict__ deg, int E) {
  const int wavesPerBlock = blockDim.x >> 5;
  const int e    = blockIdx.x * wavesPerBlock + (threadIdx.x >> 5);
  const int lane = threadIdx.x & 31;
  if (e >= E) return;
  const int r = row[e], c = col[e];
  const float* ar = a + (size_t)r * F_DIM;
  const float* ac = a + (size_t)c * F_DIM;
#pragma unroll
  for (int h = 0; h < H_DIM; ++h) {
    float v = ar[h * A_DIM + lane] * ac[h * A_DIM + lane];
#pragma unroll
    for (int off = 16; off > 0; off >>= 1)
      v += __shfl_xor(v, off, 32);
    if (lane == 0) {
      const float s = sigmoidf_(v);
      attn[(size_t)e * H_DIM + h] = s;
      atomicAdd(&deg[(size_t)r * H_DIM + h], s);
    }
  }
}

__global__ void igat_zero_f32(float* __restrict__ buf, size_t n) {
  const size_t stride = (size_t)blockDim.x * gridDim.x;
  for (size_t i = (size_t)blockIdx.x * blockDim.x + threadIdx.x; i < n; i += stride)
    buf[i] = 0.0f;
}

// deg -> inv_sqrt(deg) in place, with deg>0 guard
__global__ void igat_deginv(float* __restrict__ deg, int n) {
  const int i = blockIdx.x * blockDim.x + threadIdx.x;
  if (i < n) {
    const float d = deg[i];
    deg[i] = (d > 0.0f) ? rsqrtf(d) : 0.0f;
  }
}

__global__ void igat_vals(const int* __restrict__ row, const int* __restrict__ col,
                          const float* __restrict__ attn, const float* __restrict__ dinv,
                          float* __restrict__ vals, int E) {
  const int e = blockIdx.x * blockDim.x + threadIdx.x;
  if (e < E) {
    const int r = row[e], c = col[e];
#pragma unroll
    for (int h = 0; h < H_DIM; ++h)
      vals[(size_t)e * H_DIM + h] = attn[(size_t)e * H_DIM + h] *
                                    dinv[(size_t)r * H_DIM + h] *
                                    dinv[(size_t)c * H_DIM + h];
  }
}

// Scalars layout in sc[]: [0:4)=rs, [4:8)=rsnew, [8:12)=pAp, [12:16)=scale
__global__ void igat_scalars_init(const float* __restrict__ eps_inv, float* __restrict__ sc) {
  const int t = threadIdx.x;
  if (t < H_DIM) {
    sc[t]      = 0.0f;                            // rs
    sc[4 + t]  = 0.0f;                            // rsnew
    sc[8 + t]  = 0.0f;                            // pAp
    sc[12 + t] = 1.0f - sigmoidf_(eps_inv[t]);    // scale = 1 - eps
  }
}

// CG init: x=0, r=p=b, rs[h] = sum b^2 (per head). blockDim must be 256.
__global__ void igat_cg_init(const float* __restrict__ b, float* __restrict__ x,
                             float* __restrict__ r, float* __restrict__ p,
                             float* __restrict__ sc, size_t NT) {
  __shared__ float s[H_DIM];
  const int tid = threadIdx.x;
  const int h = tid >> 6;          // valid: base & stride are multiples of 256
  if (tid < H_DIM) s[tid] = 0.0f;
  __syncthreads();
  float partial = 0.0f;
  const size_t stride = (size_t)blockDim.x * gridDim.x;
  for (size_t i = (size_t)blockIdx.x * blockDim.x + tid; i < NT; i += stride) {
    const float bv = b[i];
    x[i] = 0.0f; r[i] = bv; p[i] = bv;
    partial += bv * bv;
  }
  atomicAdd(&s[h], partial);
  __syncthreads();
  if (tid < H_DIM) atomicAdd(&sc[tid], s[tid]);
}

// Per-iteration zero: Ap=0; block 0 rotates rs<=rsnew and clears rsnew/pAp.
__global__ void igat_iter_zero(float* __restrict__ Ap, float* __restrict__ sc,
                               int first, size_t NT) {
  const size_t stride = (size_t)blockDim.x * gridDim.x;
  for (size_t i = (size_t)blockIdx.x * blockDim.x + threadIdx.x; i < NT; i += stride)
    Ap[i] = 0.0f;
  if (blockIdx.x == 0 && threadIdx.x < H_DIM) {
    if (!first) sc[threadIdx.x] = sc[4 + threadIdx.x];  // rs = rsnew
    sc[4 + threadIdx.x] = 0.0f;                         // rsnew = 0
    sc[8 + threadIdx.x] = 0.0f;                         // pAp = 0
  }
}

// SpMM scatter: Ap[row[e], h, :] += vals[e,h] * p[col[e], h, :]
// One wave per edge, 256 floats per edge (8 per lane). p (51 MB) is
// L2-resident on MI455X (192 MB), so the random gathers never hit HBM.
__global__ void igat_spmm(const int* __restrict__ row, const int* __restrict__ col,
                          const float* __restrict__ vals, const float* __restrict__ p,
                          float* __restrict__ Ap, int E) {
  const int wavesPerBlock = blockDim.x >> 5;
  const int e    = blockIdx.x * wavesPerBlock + (threadIdx.x >> 5);
  const int lane = threadIdx.x & 31;
  if (e >= E) return;
  const int r = row[e], c = col[e];
  const float* __restrict__ pc  = p  + (size_t)c * HO;
  float* __restrict__       apr = Ap + (size_t)r * HO;
  __builtin_prefetch(pc + lane, 0, 0);   // global_prefetch_b8 warm-up of gather line
  const float4 v4 = *(const float4*)(vals + (size_t)e * H_DIM);
#pragma unroll
  for (int k = 0; k < 8; ++k) {
    const int i = lane + 32 * k;
    const int h = i >> 6;
    const float vh = (h == 0) ? v4.x : (h == 1) ? v4.y : (h == 2) ? v4.z : v4.w;
    atomicAdd(&apr[i], vh * pc[i]);
  }
}

// Finish matvec: Ap = p - scale[h]*Ap (in place); pAp[h] += <p, Ap>
__global__ void igat_finish_matvec(const float* __restrict__ p, float* __restrict__ Ap,
                                   float* __restrict__ sc, size_t NT) {
  __shared__ float s[H_DIM];
  const int tid = threadIdx.x;
  const int h = tid >> 6;
  if (tid < H_DIM) s[tid] = 0.0f;
  __syncthreads();
  const float scale = sc[12 + h];
  float partial = 0.0f;
  const size_t stride = (size_t)blockDim.x * gridDim.x;
  for (size_t i = (size_t)blockIdx.x * blockDim.x + tid; i < NT; i += stride) {
    const float pv  = p[i];
    const float apv = pv - scale * Ap[i];
    Ap[i] = apv;
    partial += pv * apv;
  }
  atomicAdd(&s[h], partial);
  __syncthreads();
  if (tid < H_DIM) atomicAdd(&sc[8 + tid], s[tid]);
}

// x += alpha*p; r -= alpha*Ap; rsnew[h] += r^2   (alpha = rs/pAp per head)
__global__ void igat_update_xr(const float* __restrict__ p, const float* __restrict__ Ap,
                               float* __restrict__ x, float* __restrict__ r,
                               float* __restrict__ sc, size_t NT) {
  __shared__ float s[H_DIM];
  const int tid = threadIdx.x;
  const int h = tid >> 6;
  if (tid < H_DIM) s[tid] = 0.0f;
  __syncthreads();
  const float rs_h  = sc[h];
  const float pAp_h = sc[8 + h];
  const float alpha = (pAp_h != 0.0f) ? rs_h / pAp_h : 0.0f;
  float partial = 0.0f;
  const size_t stride = (size_t)blockDim.x * gridDim.x;
  for (size_t i = (size_t)blockIdx.x * blockDim.x + tid; i < NT; i += stride) {
    x[i] += alpha * p[i];
    const float rv = r[i] - alpha * Ap[i];
    r[i] = rv;
    partial += rv * rv;
  }
  atomicAdd(&s[h], partial);
  __syncthreads();
  if (tid < H_DIM) atomicAdd(&sc[4 + tid], s[tid]);
}

// p = r + beta*p   (beta = rsnew/rs per head)
__global__ void igat_update_p(const float* __restrict__ r, float* __restrict__ p,
                              const float* __restrict__ sc, size_t NT) {
  const int tid = threadIdx.x;
  const int h = tid >> 6;
  const float rs_h    = sc[h];
  const float rsnew_h = sc[4 + h];
  const float beta = (rs_h != 0.0f) ? rsnew_h / rs_h : 0.0f;
  const size_t stride = (size_t)blockDim.x * gridDim.x;
  for (size_t i = (size_t)blockIdx.x * blockDim.x + tid; i < NT; i += stride)
    p[i] = r[i] + beta * p[i];
}

// out[n,o] = mean_h x[n,h,o]
__global__ void igat_head_mean(const float* __restrict__ x, float* __restrict__ out, int N) {
  const int i = blockIdx.x * blockDim.x + threadIdx.x;
  if (i < N * O_DIM) {
    const int nrow = i >> 6;
    const int o = i & 63;
    const float* xr = x + (size_t)nrow * HO + o;
    out[i] = 0.25f * (xr[0] + xr[O_DIM] + xr[2 * O_DIM] + xr[3 * O_DIM]);
  }
}

// ---------------------------------------------------------------------------
extern "C" void kernel_launch(void* const* d_in, const int* in_sizes, int n_in,
                              void* d_out, int out_size, void* d_ws, size_t ws_size,
                              hipStream_t stream) {
  (void)n_in; (void)out_size; (void)ws_size;
  const int*   row     = (const int*)d_in[0];
  const int*   col     = (const int*)d_in[1];
  const float* feat    = (const float*)d_in[2];
  const float* w_attn  = (const float*)d_in[3];
  const float* b_attn  = (const float*)d_in[4];
  const float* w_out   = (const float*)d_in[5];
  const float* b_out   = (const float*)d_in[6];
  const float* eps_inv = (const float*)d_in[7];
  float* out = (float*)d_out;

  const int E = in_sizes[0];
  const int N = in_sizes[2] / F_DIM;
  const size_t NT = (size_t)N * HO;

  // Workspace carve-up (all chunks 16B-aligned): ~303 MB total for N=50k, E=650k
  float* ws   = (float*)d_ws;
  float* sc   = ws;                          // 32 scalars
  float* a    = sc  + 32;                    // [N,128] attention features
  float* b    = a   + (size_t)N * F_DIM;     // [N,256] rhs (f)
  float* x    = b   + NT;                    // CG state
  float* r    = x   + NT;
  float* p    = r   + NT;
  float* Ap   = p   + NT;
  float* attn = Ap  + NT;                    // [E,4]
  float* vals = attn + (size_t)E * H_DIM;    // [E,4]
  float* deg  = vals + (size_t)E * H_DIM;    // [N,4] -> becomes inv_sqrt

  igat_scalars_init<<<1, 32, 0, stream>>>(eps_inv, sc);
  igat_zero_f32<<<256, 256, 0, stream>>>(deg, (size_t)N * H_DIM);

  // GEMMs via f32 WMMA: 8 waves/block, 128x16 panel per block, B tile in LDS
  const int nPanel = (N + 127) / 128;
  igat_gemm_wmma<<<dim3(nPanel, (H_DIM * A_DIM) / 16), 256, 0, stream>>>(
      feat, w_attn, b_attn, a, N, H_DIM * A_DIM);
  igat_gemm_wmma<<<dim3(nPanel, HO / 16), 256, 0, stream>>>(
      feat, w_out, b_out, b, N, HO);

  igat_edge_attn<<<(E + 7) / 8, 256, 0, stream>>>(row, col, a, attn, deg, E);
  igat_deginv<<<(N * H_DIM + 255) / 256, 256, 0, stream>>>(deg, N * H_DIM);
  igat_vals<<<(E + 255) / 256, 256, 0, stream>>>(row, col, attn, deg, vals, E);

  igat_cg_init<<<1024, 256, 0, stream>>>(b, x, r, p, sc, NT);

  for (int it = 0; it < CG_ITERS; ++it) {
    igat_iter_zero<<<2048, 256, 0, stream>>>(Ap, sc, it == 0 ? 1 : 0, NT);
    igat_spmm<<<(E + 7) / 8, 256, 0, stream>>>(row, col, vals, p, Ap, E);
    igat_finish_matvec<<<1024, 256, 0, stream>>>(p, Ap, sc, NT);
    igat_update_xr<<<1024, 256, 0, stream>>>(p, Ap, x, r, sc, NT);
    igat_update_p<<<1024, 256, 0, stream>>>(r, p, sc, NT);
  }

  igat_head_mean<<<(N * O_DIM + 255) / 256, 256, 0, stream>>>(x, out, N);
}